// MyChamferDistance_40888088658143
// MI455X (gfx1250) — compile-verified
//
#include <hip/hip_runtime.h>
#include <hip/hip_bf16.h>

typedef float v2f __attribute__((ext_vector_type(2)));
typedef float v8f __attribute__((ext_vector_type(8)));

#define B_  8
#define N_  4096
#define M_  4096
// D = 2 (encoded into the K=4 augmented WMMA trick)

#define ROWS_PER_WAVE  32            // two 16-row WMMA strips per wave
#define ROWS_PER_BLOCK 256           // 8 waves * 32 rows
#define BLOCKS_PER_BATCH (N_ / ROWS_PER_BLOCK)   // 16
#define MTILES (M_ / 16)                          // 256
#define LDS_BYTES (2 * M_ * sizeof(float2) + M_ * sizeof(unsigned))  // 64KB + 16KB

// Raw hardware min/max: IEEE minimumNumber semantics, no canonicalization.
__device__ __forceinline__ float vmin(float a, float b) {
    float d;
    asm("v_min_num_f32 %0, %1, %2" : "=v"(d) : "v"(a), "v"(b));
    return d;
}
__device__ __forceinline__ float vmin3(float a, float b, float c) {
    float d;
    asm("v_min3_num_f32 %0, %1, %2, %3" : "=v"(d) : "v"(a), "v"(b), "v"(c));
    return d;
}
__device__ __forceinline__ float vmax0(float a) {
    float d;
    asm("v_max_num_f32 %0, 0, %1" : "=v"(d) : "v"(a));
    return d;
}

// ---------------------------------------------------------------------------
// ws layout: [0, B*M*4)           : minM as uint bit-patterns (global min d^2)
//            [B*M*4, B*M*4+B*N*4) : minN as float d^2 (exclusive stores)
// ---------------------------------------------------------------------------

__global__ void chamfer_init_kernel(unsigned* __restrict__ minM) {
    int i = blockIdx.x * blockDim.x + threadIdx.x;
    if (i < B_ * M_) minM[i] = 0x7F800000u;  // +inf
}

__global__ __launch_bounds__(256)
void chamfer_tiles_kernel(const float* __restrict__ x,
                          const float* __restrict__ tgt,
                          unsigned* __restrict__ minM,
                          float* __restrict__ minN) {
    extern __shared__ __align__(16) char smem[];
    float2*   bOp    = (float2*)smem;                               // [2*M_]
    unsigned* colMin = (unsigned*)(smem + 2 * M_ * sizeof(float2)); // [M_]

    const int b      = blockIdx.x / BLOCKS_PER_BATCH;
    const int nChunk = blockIdx.x % BLOCKS_PER_BATCH;
    const int tid    = threadIdx.x;
    const int lane   = tid & 31;
    const int wv     = tid >> 5;

    // --- Preprocess target points for this batch into LDS (once per block) --
    const float* tb = tgt + (size_t)b * M_ * 2;
    for (int j = tid; j < M_; j += 256) {
        float tx = tb[2 * j], ty = tb[2 * j + 1];
        bOp[j]      = make_float2(-2.0f * tx, -2.0f * ty);  // B rows K0,K1
        bOp[M_ + j] = make_float2(1.0f, tx * tx + ty * ty); // B rows K2,K3
        colMin[j]   = 0x7F800000u;
    }
    __syncthreads();

    // --- Build two A operands (rows nBase..+15 and nBase+16..+31) -----------
    // ISA 16x4 f32 A layout: lanes 0-15 hold (K0,K1), lanes 16-31 hold (K2,K3)
    const int r     = lane & 15;
    const int half  = lane >> 4;
    const int nBase = nChunk * ROWS_PER_BLOCK + wv * ROWS_PER_WAVE;
    const float* xb = x + (size_t)b * N_ * 2;

    v2f a0, a1;
    {
        float x0 = xb[2 * (nBase + r)],      y0 = xb[2 * (nBase + r) + 1];
        float x1 = xb[2 * (nBase + 16 + r)], y1 = xb[2 * (nBase + 16 + r) + 1];
        if (half == 0) { a0[0] = x0; a0[1] = y0; a1[0] = x1; a1[1] = y1; }
        else {
            a0[0] = x0 * x0 + y0 * y0; a0[1] = 1.0f;   // (||x||^2, 1)
            a1[0] = x1 * x1 + y1 * y1; a1[1] = 1.0f;
        }
    }

    float rowAcc0[8], rowAcc1[8];
#pragma unroll
    for (int j = 0; j < 8; ++j) { rowAcc0[j] = 3.0e38f; rowAcc1[j] = 3.0e38f; }

    // --- Main loop (x2 unroll): 2 LDS loads feed 4 WMMAs -------------------
    //     64 rows*cols... : 32 rows x 32 cols of d^2 per iteration
    const float2* bLane = bOp + half * M_ + r;    // indexed by tile*16
    const int tstart = (wv * 32) & (MTILES - 1);  // stagger waves
    float2 bvA = bLane[16 * tstart];
    float2 bvB = bLane[16 * ((tstart + 1) & (MTILES - 1))];

    for (int i = 0; i < MTILES; i += 2) {
        const int tA = (i + wv * 32) & (MTILES - 1);
        const int tB = (i + 1 + wv * 32) & (MTILES - 1);
        v2f bbA, bbB;
        bbA[0] = bvA.x; bbA[1] = bvA.y;
        bbB[0] = bvB.x; bbB[1] = bvB.y;
        // prefetch the next two tiles (consumed next iteration)
        bvA = bLane[16 * ((i + 2 + wv * 32) & (MTILES - 1))];
        bvB = bLane[16 * ((i + 3 + wv * 32) & (MTILES - 1))];

        v8f c0A = {}, c1A = {}, c0B = {}, c1B = {};
        c0A = __builtin_amdgcn_wmma_f32_16x16x4_f32(
                false, a0, false, bbA, (short)0, c0A, false, false);
        c1A = __builtin_amdgcn_wmma_f32_16x16x4_f32(
                false, a1, false, bbA, (short)0, c1A, false, false);
        c0B = __builtin_amdgcn_wmma_f32_16x16x4_f32(
                false, a0, false, bbB, (short)0, c0B, false, false);
        c1B = __builtin_amdgcn_wmma_f32_16x16x4_f32(
                false, a1, false, bbB, (short)0, c1B, false, false);

        // row mins: one min3 absorbs both tiles' contributions (raw d^2,
        // clamp deferred: max(.,0) is monotone so it commutes with min)
#pragma unroll
        for (int j = 0; j < 8; ++j) rowAcc0[j] = vmin3(rowAcc0[j], c0A[j], c0B[j]);
#pragma unroll
        for (int j = 0; j < 8; ++j) rowAcc1[j] = vmin3(rowAcc1[j], c1A[j], c1B[j]);

        // column partials: min over this lane's 16 rows (8 from each strip)
        float cpA = vmin3(c0A[0], c1A[0], c0A[1]);
        cpA = vmin3(cpA, c1A[1], c0A[2]);
        cpA = vmin3(cpA, c1A[2], c0A[3]);
        cpA = vmin3(cpA, c1A[3], c0A[4]);
        cpA = vmin3(cpA, c1A[4], c0A[5]);
        cpA = vmin3(cpA, c1A[5], c0A[6]);
        cpA = vmin3(cpA, c1A[6], c0A[7]);
        cpA = vmin(cpA, c1A[7]);
        atomicMin(&colMin[tA * 16 + r], __float_as_uint(vmax0(cpA)));

        float cpB = vmin3(c0B[0], c1B[0], c0B[1]);
        cpB = vmin3(cpB, c1B[1], c0B[2]);
        cpB = vmin3(cpB, c1B[2], c0B[3]);
        cpB = vmin3(cpB, c1B[3], c0B[4]);
        cpB = vmin3(cpB, c1B[4], c0B[5]);
        cpB = vmin3(cpB, c1B[5], c0B[6]);
        cpB = vmin3(cpB, c1B[6], c0B[7]);
        cpB = vmin(cpB, c1B[7]);
        atomicMin(&colMin[tB * 16 + r], __float_as_uint(vmax0(cpB)));
    }

    // --- Finalize row mins: reduce across the 16 lanes of each half ---------
#pragma unroll
    for (int j = 0; j < 8; ++j) {
        float v0 = rowAcc0[j], v1 = rowAcc1[j];
        v0 = vmin(v0, __shfl_xor(v0, 1, 32));
        v1 = vmin(v1, __shfl_xor(v1, 1, 32));
        v0 = vmin(v0, __shfl_xor(v0, 2, 32));
        v1 = vmin(v1, __shfl_xor(v1, 2, 32));
        v0 = vmin(v0, __shfl_xor(v0, 4, 32));
        v1 = vmin(v1, __shfl_xor(v1, 4, 32));
        v0 = vmin(v0, __shfl_xor(v0, 8, 32));
        v1 = vmin(v1, __shfl_xor(v1, 8, 32));
        rowAcc0[j] = vmax0(v0);                   // clamp once at the end
        rowAcc1[j] = vmax0(v1);
    }
    if (lane == 0) {
#pragma unroll
        for (int j = 0; j < 8; ++j) {
            minN[(size_t)b * N_ + nBase + j]      = rowAcc0[j];
            minN[(size_t)b * N_ + nBase + 16 + j] = rowAcc1[j];
        }
    } else if (lane == 16) {
#pragma unroll
        for (int j = 0; j < 8; ++j) {
            minN[(size_t)b * N_ + nBase + 8 + j]  = rowAcc0[j];
            minN[(size_t)b * N_ + nBase + 24 + j] = rowAcc1[j];
        }
    }

    // --- Merge this block's column mins into the global per-batch array -----
    __syncthreads();
    for (int j = tid; j < M_; j += 256)
        atomicMin(&minM[(size_t)b * M_ + j], colMin[j]);   // global_atomic_min_u32
}

__global__ __launch_bounds__(1024)
void chamfer_final_kernel(const unsigned* __restrict__ minM,
                          const float* __restrict__ minN,
                          float* __restrict__ out) {
    __shared__ float s1[1024];
    __shared__ float s2[1024];
    const int tid = threadIdx.x;
    float a1 = 0.0f, a2 = 0.0f;
    for (int i = tid; i < B_ * M_; i += 1024)
        a1 += sqrtf(fmaxf(__uint_as_float(minM[i]), 0.0f));
    for (int i = tid; i < B_ * N_; i += 1024)
        a2 += sqrtf(fmaxf(minN[i], 0.0f));
    s1[tid] = a1; s2[tid] = a2;
    __syncthreads();
    for (int s = 512; s > 0; s >>= 1) {      // deterministic tree reduction
        if (tid < s) { s1[tid] += s1[tid + s]; s2[tid] += s2[tid + s]; }
        __syncthreads();
    }
    if (tid == 0)
        out[0] = s1[0] / (float)(B_ * M_) + s2[0] / (float)(B_ * N_);
}

extern "C" void kernel_launch(void* const* d_in, const int* in_sizes, int n_in,
                              void* d_out, int out_size, void* d_ws, size_t ws_size,
                              hipStream_t stream) {
    const float* x   = (const float*)d_in[0];   // [B, N, 2] f32
    const float* tgt = (const float*)d_in[1];   // [B, M, 2] f32
    unsigned* minM = (unsigned*)d_ws;                                   // B*M uints
    float*    minN = (float*)((char*)d_ws + (size_t)B_ * M_ * sizeof(unsigned));
    float*    out  = (float*)d_out;

    chamfer_init_kernel<<<(B_ * M_ + 255) / 256, 256, 0, stream>>>(minM);
    chamfer_tiles_kernel<<<B_ * BLOCKS_PER_BATCH, 256, LDS_BYTES, stream>>>(
        x, tgt, minM, minN);
    chamfer_final_kernel<<<1, 1024, 0, stream>>>(minM, minN, out);
}